// myLSTM_39934605918491
// MI455X (gfx1250) — compile-verified
//
#include <hip/hip_runtime.h>
#include <hip/hip_bf16.h>

typedef __attribute__((ext_vector_type(2))) float v2f;
typedef __attribute__((ext_vector_type(8))) float v8f;

namespace {
constexpr int B  = 256;
constexpr int T  = 512;
constexpr int E  = 512;
constexpr int H  = 512;
constexpr int NC = 1000;
constexpr int KTOT = E + H;   // 1024
constexpr int KC   = 64;      // K chunk staged in LDS (E is a multiple -> uniform A source)
constexpr int KP   = KC + 4;  // padded row stride: 8B-aligned rows + conflict-free frag loads
constexpr int ROWS = 32;      // batch rows per block (2 wave row-groups)
constexpr int COLS = 64;      // H columns per block (2 wave col-groups x 32)
constexpr int NT   = 128;     // 4 waves, arranged 2 (M) x 2 (N); each wave: 16x32, dual acc
}

// One LSTM timestep: h_new = sigmoid(g2)*tanh( sigmoid(g0)*tanh(xh@cell_W + cell_b) )
// (cell state c stays identically zero in the reference, so the gf*c term vanishes)
__global__ __launch_bounds__(NT) void lstm_step_kernel(
    const int* __restrict__ tokens, const float* __restrict__ embed,
    const float* __restrict__ gate_W, const float* __restrict__ gate_b,
    const float* __restrict__ cell_W, const float* __restrict__ cell_b,
    const float* __restrict__ h_old, float* __restrict__ h_new, int t)
{
    __shared__ float As [ROWS][KP];      // xh tile, K-contiguous per row
    __shared__ float BsT[COLS][KP];      // cell_W tile TRANSPOSED: [col][k], K-contiguous
    __shared__ float Gs [ROWS][4];       // sigmoid gates per batch row

    const int tid  = threadIdx.x;
    const int wave = tid >> 5;
    const int lane = tid & 31;
    const int m    = lane & 15;
    const int hi   = lane >> 4;
    const int mw   = wave >> 1;          // wave row-group (0..1)
    const int nw   = wave & 1;           // wave col-group (0..1)

    const int row0 = blockIdx.y * ROWS;
    const int col0 = blockIdx.x * COLS;
    const int cA   = nw * 32 + m;        // local column of acc0 (acc1 = +16)
    const int am   = mw * 16 + m;        // local A row for this lane

    const int grow = tid / 3, gj = tid % 3;   // gate work (tid < 96)
    float gacc = 0.0f;
    v8f acc0 = {}, acc1 = {};

    for (int k0 = 0; k0 < KTOT; k0 += KC) {
        const bool inEmb = (k0 < E);     // uniform: chunk never straddles the boundary

        // Stage A tile (float4): xh[row][k] from embed gather or h_old
        #pragma unroll
        for (int i = tid; i < ROWS * (KC / 4); i += NT) {
            const int r  = i >> 4;               // KC/4 == 16
            const int c4 = (i & 15) << 2;
            const int bb = row0 + r;
            const float* src = inEmb
                ? (embed + (size_t)tokens[bb * T + t] * E + (k0 + c4))
                : (h_old + (size_t)bb * H + (k0 - E + c4));
            const float4 v = *(const float4*)src;
            As[r][c4 + 0] = v.x; As[r][c4 + 1] = v.y;
            As[r][c4 + 2] = v.z; As[r][c4 + 3] = v.w;
        }
        // Stage B tile transposed: BsT[col][k] = cell_W[k0+k][col0+col]
        #pragma unroll
        for (int i = tid; i < KC * (COLS / 4); i += NT) {
            const int r  = i >> 4;               // k row within chunk (COLS/4 == 16)
            const int c4 = (i & 15) << 2;        // first of 4 columns
            const float4 v = *(const float4*)(cell_W + (size_t)(k0 + r) * H + col0 + c4);
            BsT[c4 + 0][r] = v.x; BsT[c4 + 1][r] = v.y;
            BsT[c4 + 2][r] = v.z; BsT[c4 + 3][r] = v.w;
        }
        __syncthreads();

        // Fused gate GEMM partials: (32 rows x 3 gates) over this chunk
        if (tid < ROWS * 3) {
            float s = gacc;
            for (int c = 0; c < KC; ++c)
                s += As[grow][c] * gate_W[(size_t)(k0 + c) * 3 + gj];
            gacc = s;
        }

        // Software-pipelined WMMA: every fragment is a contiguous 8B-aligned LDS pair
        v2f a, b0, b1;
        {
            const int ka = 2 * hi;
            a.x  = As [am     ][ka];  a.y  = As [am     ][ka + 1];
            b0.x = BsT[cA     ][ka];  b0.y = BsT[cA     ][ka + 1];
            b1.x = BsT[cA + 16][ka];  b1.y = BsT[cA + 16][ka + 1];
        }
        #pragma unroll
        for (int kk = 0; kk < KC; kk += 4) {
            v2f an = {}, b0n = {}, b1n = {};
            if (kk + 4 < KC) {
                const int ka = kk + 4 + 2 * hi;
                an.x  = As [am     ][ka];  an.y  = As [am     ][ka + 1];
                b0n.x = BsT[cA     ][ka];  b0n.y = BsT[cA     ][ka + 1];
                b1n.x = BsT[cA + 16][ka];  b1n.y = BsT[cA + 16][ka + 1];
            }
            acc0 = __builtin_amdgcn_wmma_f32_16x16x4_f32(
                false, a, false, b0, (short)0, acc0, false, false);
            acc1 = __builtin_amdgcn_wmma_f32_16x16x4_f32(
                false, a, false, b1, (short)0, acc1, false, false);
            a = an; b0 = b0n; b1 = b1n;
        }
        __syncthreads();
    }

    if (tid < ROWS * 3)
        Gs[grow][gj] = 1.0f / (1.0f + __expf(-(gacc + gate_b[gj])));
    __syncthreads();

    const int colg0 = col0 + cA;
    const int colg1 = colg0 + 16;
    const float cb0 = cell_b[colg0];
    const float cb1 = cell_b[colg1];
    #pragma unroll
    for (int r = 0; r < 8; ++r) {
        const int mr = mw * 16 + r + 8 * hi;     // C/D layout: M = r + 8*(lane/16)
        const int bb = row0 + mr;
        const float gi = Gs[mr][0];
        const float go = Gs[mr][2];
        const float C0 = gi * tanhf(acc0[r] + cb0);   // gf*c == 0
        const float C1 = gi * tanhf(acc1[r] + cb1);
        h_new[(size_t)bb * H + colg0] = go * tanhf(C0);
        h_new[(size_t)bb * H + colg1] = go * tanhf(C1);
    }
}

// Output projection: out = h_T @ out_W + out_b   (256x512)@(512x1000)
__global__ __launch_bounds__(NT) void lstm_out_kernel(
    const float* __restrict__ h, const float* __restrict__ out_W,
    const float* __restrict__ out_b, float* __restrict__ out)
{
    const int tid  = threadIdx.x;
    const int wave = tid >> 5;
    const int lane = tid & 31;
    const int m    = lane & 15;
    const int hi   = lane >> 4;

    const int row0 = blockIdx.y * 16;
    const int col  = blockIdx.x * 64 + wave * 16 + m;

    v8f acc = {};
    for (int k = 0; k < H; k += 4) {
        const int ka = k + 2 * hi;
        v2f a, b;
        a.x = h[(size_t)(row0 + m) * H + ka];
        a.y = h[(size_t)(row0 + m) * H + ka + 1];
        b.x = (col < NC) ? out_W[(size_t)ka * NC + col] : 0.0f;
        b.y = (col < NC) ? out_W[(size_t)(ka + 1) * NC + col] : 0.0f;
        acc = __builtin_amdgcn_wmma_f32_16x16x4_f32(
            false, a, false, b, (short)0, acc, false, false);
    }
    if (col < NC) {
        const float ob = out_b[col];
        #pragma unroll
        for (int r = 0; r < 8; ++r) {
            const int mr = r + 8 * hi;
            out[(size_t)(row0 + mr) * NC + col] = acc[r] + ob;
        }
    }
}

__global__ void zero_f32_kernel(float* __restrict__ p, int n)
{
    const int i = blockIdx.x * blockDim.x + threadIdx.x;
    if (i < n) p[i] = 0.0f;
}

extern "C" void kernel_launch(void* const* d_in, const int* in_sizes, int n_in,
                              void* d_out, int out_size, void* d_ws, size_t ws_size,
                              hipStream_t stream)
{
    const int*   tokens = (const int*)  d_in[0];
    const float* embed  = (const float*)d_in[1];
    const float* gate_W = (const float*)d_in[2];
    const float* gate_b = (const float*)d_in[3];
    const float* cell_W = (const float*)d_in[4];
    const float* cell_b = (const float*)d_in[5];
    const float* out_W  = (const float*)d_in[6];
    const float* out_b  = (const float*)d_in[7];
    float*       out    = (float*)d_out;

    float* hbuf0 = (float*)d_ws;               // B*H floats
    float* hbuf1 = hbuf0 + (size_t)B * H;      // B*H floats (double buffer, 1 MB total)

    zero_f32_kernel<<<(B * H + 255) / 256, 256, 0, stream>>>(hbuf0, B * H);

    dim3 gridStep(H / COLS, B / ROWS);         // 8 x 8 = 64 blocks per step
    for (int t = 0; t < T; ++t) {
        const float* hin  = (t & 1) ? hbuf1 : hbuf0;
        float*       hout = (t & 1) ? hbuf0 : hbuf1;
        lstm_step_kernel<<<gridStep, NT, 0, stream>>>(
            tokens, embed, gate_W, gate_b, cell_W, cell_b, hin, hout, t);
    }
    // T even: final h lives in hbuf0 (t=511 wrote hbuf0)
    const float* hfin = hbuf0;

    dim3 gridOut((NC + 63) / 64, B / 16);      // 16 x 16 blocks, ragged N guarded
    lstm_out_kernel<<<gridOut, NT, 0, stream>>>(hfin, out_W, out_b, out);
}